// GCLayer_10514079941180
// MI455X (gfx1250) — compile-verified
//
#include <hip/hip_runtime.h>

typedef __attribute__((ext_vector_type(2))) float v2f;
typedef __attribute__((ext_vector_type(8))) float v8f;

#define B_    32
#define CIN_  64
#define COUT_ 64
#define IMG_  128
#define PIX_  (IMG_ * IMG_)   // 16384

// ---------------------------------------------------------------------------
// Phase 1: xs[b,h,w] = sum_c x[b,c,h,w].  Pure bandwidth: 134 MB in, 2 MB out.
// Each thread owns 4 consecutive w (float4), loops the 64 channels with a
// 64KB stride; loads are fully coalesced across the wave for every channel.
// ---------------------------------------------------------------------------
__global__ __launch_bounds__(256) void gc_chansum(const float* __restrict__ x,
                                                  float* __restrict__ xs) {
    int t  = blockIdx.x * blockDim.x + threadIdx.x;  // 0..131071
    int p0 = t << 2;                                 // pixel group base
    int b  = p0 >> 14;                               // /16384
    int hw = p0 & (PIX_ - 1);
    const float4* xp = (const float4*)(x + (size_t)b * CIN_ * PIX_ + hw);
    float4 acc = make_float4(0.f, 0.f, 0.f, 0.f);
#pragma unroll
    for (int c = 0; c < CIN_; ++c) {
        float4 v = xp[c * (PIX_ / 4)];
        acc.x += v.x; acc.y += v.y; acc.z += v.z; acc.w += v.w;
    }
    *(float4*)(xs + (size_t)b * PIX_ + hw) = acc;
}

// ---------------------------------------------------------------------------
// Phase 2: 3x3 conv as GEMM  out[o,pix] = sum_t K[o,t] * patch[t,pix]
// via V_WMMA_F32_16X16X4_F32.  One wave = 16 outputs x 16 pixels (one row
// strip), K = 9 taps padded to 12 -> 3 WMMA k-steps.
//   A 16x4 f32 layout : lane L (M = L%16): v0 holds K = 2*(L>=16), v1 = K+1
//   B  4x16 f32 layout: mirrored (N = L%16)
//   C/D 16x16 f32     : row r, lane L -> M = r + 8*(L>=16), N = L%16
// Zero padding handled with clamped addresses + selects (no divergence, EXEC
// all-1s at each WMMA per ISA requirement).
// ---------------------------------------------------------------------------
__global__ __launch_bounds__(256) void gc_conv_wmma(const float* __restrict__ xs,
                                                    const float* __restrict__ kern,  // [64][9]
                                                    const float* __restrict__ bias,  // [64]
                                                    float* __restrict__ out) {
    const int lane = threadIdx.x & 31;
    const int wave = (blockIdx.x << 3) + (threadIdx.x >> 5);  // 0..131071
    const int wt = wave & 7;            // w tile (16 px)
    const int h  = (wave >> 3) & 127;   // row
    const int ot = (wave >> 10) & 3;    // output-channel tile (16 ch)
    const int b  = wave >> 12;          // batch

    const int n    = lane & 15;         // N (pixel) and M (out chan) sub-index
    const int half = lane >> 4;         // 0 / 1
    const int o    = ot * 16 + n;       // A-matrix row (output channel)
    const int w0   = wt * 16;
    const float* xsb = xs + (size_t)b * PIX_;

    v8f c = {};
#pragma unroll
    for (int kc = 0; kc < 3; ++kc) {
        const int t0 = kc * 4 + 2 * half;   // lane-local global tap index

        // A operand: weights for taps t0, t0+1 (zero past tap 8)
        v2f a;
        {
            int ta = t0;
            float av = kern[o * 9 + (ta < 9 ? ta : 8)];
            a.x = (ta < 9) ? av : 0.0f;
            ta = t0 + 1;
            av = kern[o * 9 + (ta < 9 ? ta : 8)];
            a.y = (ta < 9) ? av : 0.0f;
        }

        // B operand: xs patch values for pixel n at taps t0, t0+1
        v2f bm;
#pragma unroll
        for (int j = 0; j < 2; ++j) {
            const int t  = t0 + j;
            const int dh = t / 3 - 1;
            const int dw = t - (t / 3) * 3 - 1;
            const int hh = h + dh;
            const int ww = w0 + n + dw;
            const bool ok = (t < 9) & ((unsigned)hh < 128u) & ((unsigned)ww < 128u);
            const int hc = hh < 0 ? 0 : (hh > 127 ? 127 : hh);
            const int wc = ww < 0 ? 0 : (ww > 127 ? 127 : ww);
            const float v = xsb[hc * IMG_ + wc];   // always-valid address
            ((float*)&bm)[j] = ok ? v : 0.0f;      // select, not branch
        }

        // D = A*B + C   (8 args: neg_a, A, neg_b, B, c_mod, C, reuse_a, reuse_b)
        c = __builtin_amdgcn_wmma_f32_16x16x4_f32(false, a, false, bm,
                                                  (short)0, c, false, false);
    }

    // Add cin*bias and store the 16x16 tile (8 rows per lane per layout).
    const int obase = ot * 16 + 8 * half;
    float* op = out + (size_t)b * COUT_ * PIX_ + (size_t)h * IMG_ + w0 + n;
#pragma unroll
    for (int r = 0; r < 8; ++r) {
        float val = c[r] + (float)CIN_ * bias[obase + r];
        op[(size_t)(obase + r) * PIX_] = val;
    }
}

// ---------------------------------------------------------------------------
extern "C" void kernel_launch(void* const* d_in, const int* in_sizes, int n_in,
                              void* d_out, int out_size, void* d_ws, size_t ws_size,
                              hipStream_t stream) {
    const float* x    = (const float*)d_in[0];   // (32,64,128,128) f32
    const float* kern = (const float*)d_in[1];   // (64,3,3) f32
    const float* bias = (const float*)d_in[2];   // (64,1,1,1) f32
    float* out = (float*)d_out;                  // (32,64,128,128) f32
    float* xs  = (float*)d_ws;                   // needs 32*128*128*4 = 2 MB

    // Phase 1: channel reduction. 131072 threads, 4 pixels each.
    gc_chansum<<<512, 256, 0, stream>>>(x, xs);

    // Phase 2: WMMA conv. 131072 waves = 16384 blocks * 8 waves.
    gc_conv_wmma<<<16384, 256, 0, stream>>>(xs, kern, bias, out);
}